// GlobalAttentionPooling_64905545777761
// MI455X (gfx1250) — compile-verified
//
#include <hip/hip_runtime.h>
#include <hip/hip_bf16.h>
#include <stdint.h>

// ---------------------------------------------------------------------------
// GlobalAttentionPooling on MI455X (gfx1250), f32, bandwidth-bound.
// out[b] = (sum_n alpha_n * feat_n) @ W_feat + b_feat   (linearity of h in feat)
// ---------------------------------------------------------------------------

typedef __attribute__((ext_vector_type(2))) float v2f;
typedef __attribute__((ext_vector_type(8))) float v8f;

#define TPB 256

// ---- Kernel 0: segment start offsets via binary search (segment_ids sorted)
__global__ void k_segbounds(const int* __restrict__ seg_ids,
                            int* __restrict__ seg_start, int N, int B) {
    int b = blockIdx.x * blockDim.x + threadIdx.x;
    if (b > B) return;
    if (b == B) { seg_start[b] = N; return; }
    int lo = 0, hi = N;
    while (lo < hi) {
        int mid = (lo + hi) >> 1;
        if (seg_ids[mid] < b) lo = mid + 1; else hi = mid;
    }
    seg_start[b] = lo;
}

// ---- Kernel 1: gate[n] = dot(feat[n,:], W_gate) + b_gate  (4 lanes per row)
__global__ __launch_bounds__(TPB) void k_gate(const float* __restrict__ feat,
                                              const float* __restrict__ W_gate,
                                              const float* __restrict__ b_gate,
                                              float* __restrict__ gate, int N) {
    __shared__ float wg[128];
    int tid = threadIdx.x;
    if (tid < 128) wg[tid] = W_gate[tid];
    __syncthreads();
    int lane4 = tid & 3;
    int row = blockIdx.x * (TPB / 4) + (tid >> 2);
    if (row >= N) return;
    const float4* fp = (const float4*)(feat + (size_t)row * 128 + lane4 * 32);
    float s = 0.f;
#pragma unroll
    for (int j = 0; j < 8; ++j) {
        float4 v = fp[j];
        int k = lane4 * 32 + j * 4;
        s += v.x * wg[k] + v.y * wg[k + 1] + v.z * wg[k + 2] + v.w * wg[k + 3];
    }
    // reduce across the 4 lanes of this row (within a wave32)
    s += __shfl_xor(s, 1, 4);
    s += __shfl_xor(s, 2, 4);
    if (lane4 == 0) gate[row] = s + b_gate[0];
}

// ---- Kernel 2: one block per segment: max, sum(exp), alpha-weighted pooling
__global__ __launch_bounds__(TPB) void k_pool(const float* __restrict__ feat,
                                              const float* __restrict__ gate,
                                              const int* __restrict__ seg_start,
                                              float* __restrict__ pooled,
                                              float* __restrict__ bias_scale) {
    int b = blockIdx.x;
    int s = seg_start[b], e = seg_start[b + 1];
    int tid = threadIdx.x;
    __shared__ float red[TPB];
    __shared__ float wsh[TPB];
    __shared__ float mZ[2];

    // --- segment max of gate
    float m = -3.402823466e38f;
    for (int i = s + tid; i < e; i += TPB) m = fmaxf(m, gate[i]);
    red[tid] = m; __syncthreads();
    for (int o = TPB / 2; o > 0; o >>= 1) {
        if (tid < o) red[tid] = fmaxf(red[tid], red[tid + o]);
        __syncthreads();
    }
    if (tid == 0) mZ[0] = red[0];
    __syncthreads();
    m = mZ[0];

    // --- segment sum of exp(gate - m)
    float z = 0.f;
    for (int i = s + tid; i < e; i += TPB) z += expf(gate[i] - m);
    red[tid] = z; __syncthreads();
    for (int o = TPB / 2; o > 0; o >>= 1) {
        if (tid < o) red[tid] += red[tid + o];
        __syncthreads();
    }
    if (tid == 0) mZ[1] = red[0];
    __syncthreads();
    float invZ = (e > s && mZ[1] > 0.f) ? 1.f / mZ[1] : 0.f;

    // --- pooled[b][d] = sum_n alpha_n * feat[n][d]
    int d = tid & 127;        // feature column
    int g = tid >> 7;         // row group {0,1}
    float acc = 0.f;
    for (int base = s; base < e; base += TPB) {
        int cl = min(TPB, e - base);
        wsh[tid] = (tid < cl) ? expf(gate[base + tid] - m) * invZ : 0.f;
        __syncthreads();
        for (int r = g; r < cl; r += 2) {
            if (r + 32 < cl)  // keep HBM pipe fed: gfx1250 global_prefetch_b8
                __builtin_prefetch(&feat[(size_t)(base + r + 32) * 128 + d], 0, 0);
            acc += wsh[r] * feat[(size_t)(base + r) * 128 + d];
        }
        __syncthreads();
    }
    red[tid] = acc; __syncthreads();
    if (g == 0) pooled[b * 128 + d] = red[d] + red[d + 128];
    if (tid == 0) bias_scale[b] = (e > s) ? 1.f : 0.f;  // empty segment -> out = 0
}

// ---- Kernel 3: out[B,128] = pooled[B,128] @ W_feat[128,128] + scale*b_feat
//      f32 WMMA 16x16x4; 8 waves/block, 1 col-tile per wave, 16 rows per block.
__global__ __launch_bounds__(TPB) void k_gemm(const float* __restrict__ pooled,
                                              const float* __restrict__ W_feat,
                                              const float* __restrict__ b_feat,
                                              const float* __restrict__ bias_scale,
                                              float* __restrict__ out, int B) {
    __shared__ float As[16][132];   // padded stride: avoids LDS bank conflicts
    __shared__ float Bs[32][128];   // one 32-row K-chunk of W_feat
    int tid = threadIdx.x;
    int row0 = blockIdx.x * 16;

    // load A tile (16 x 128): 2 float4 per thread
    {
        int r = tid >> 4;
        int c = (tid & 15) * 8;
        int rs = min(row0 + r, B - 1);
        const float4* src = (const float4*)(pooled + (size_t)rs * 128 + c);
        float4 v0 = src[0], v1 = src[1];
        As[r][c + 0] = v0.x; As[r][c + 1] = v0.y; As[r][c + 2] = v0.z; As[r][c + 3] = v0.w;
        As[r][c + 4] = v1.x; As[r][c + 5] = v1.y; As[r][c + 6] = v1.z; As[r][c + 7] = v1.w;
    }

    int lane = tid & 31, wave = tid >> 5;
    int half = lane >> 4;      // ISA layout: lanes 16-31 carry K+2
    int idx  = lane & 15;      // A: row M; B/D: column N
    int col  = wave * 16 + idx;

    v8f acc = {};
    for (int kc = 0; kc < 128; kc += 32) {
        // refill Bs with W_feat rows [kc, kc+32): 4 float4 per thread
        {
            int r = tid >> 3;
            int c = (tid & 7) * 16;
            const float4* src = (const float4*)(W_feat + (size_t)(kc + r) * 128 + c);
            float4* dst = (float4*)&Bs[r][c];
            dst[0] = src[0]; dst[1] = src[1]; dst[2] = src[2]; dst[3] = src[3];
        }
        __syncthreads();
#pragma unroll
        for (int k0 = 0; k0 < 32; k0 += 4) {
            int kk = k0 + half * 2;
            v2f a  = { As[idx][kc + kk], As[idx][kc + kk + 1] };       // A[M=idx][K]
            v2f bb = { Bs[kk][col],      Bs[kk + 1][col] };            // B[K][N=col]
            acc = __builtin_amdgcn_wmma_f32_16x16x4_f32(
                false, a, false, bb, (short)0, acc, false, false);
        }
        __syncthreads();
    }

    float bias = b_feat[col];
#pragma unroll
    for (int r = 0; r < 8; ++r) {
        int grow = row0 + r + half * 8;   // C/D layout: VGPR r holds rows r, r+8
        if (grow < B)
            out[(size_t)grow * 128 + col] = acc[r] + bias_scale[grow] * bias;
    }
}

extern "C" void kernel_launch(void* const* d_in, const int* in_sizes, int n_in,
                              void* d_out, int out_size, void* d_ws, size_t ws_size,
                              hipStream_t stream) {
    const float* feat   = (const float*)d_in[0];
    const float* W_gate = (const float*)d_in[1];
    const float* b_gate = (const float*)d_in[2];
    const float* W_feat = (const float*)d_in[3];
    const float* b_feat = (const float*)d_in[4];
    const int*   segids = (const int*)d_in[5];
    int N = in_sizes[5];         // number of nodes
    int B = out_size / 128;      // number of segments
    float* out = (float*)d_out;

    // workspace layout (256B-aligned sections for float4 loads)
    uintptr_t p = (uintptr_t)d_ws;
    float* gate = (float*)p;            p += (size_t)N * 4;        p = (p + 255) & ~(uintptr_t)255;
    int* seg_start = (int*)p;           p += (size_t)(B + 1) * 4;  p = (p + 255) & ~(uintptr_t)255;
    float* bias_scale = (float*)p;      p += (size_t)B * 4;        p = (p + 255) & ~(uintptr_t)255;
    float* pooled = (float*)p;

    k_segbounds<<<(B + TPB) / TPB, TPB, 0, stream>>>(segids, seg_start, N, B);
    k_gate<<<(N + (TPB / 4) - 1) / (TPB / 4), TPB, 0, stream>>>(feat, W_gate, b_gate, gate, N);
    k_pool<<<B, TPB, 0, stream>>>(feat, gate, seg_start, pooled, bias_scale);
    k_gemm<<<(B + 15) / 16, TPB, 0, stream>>>(pooled, W_feat, b_feat, bias_scale, out, B);
}